// GCNLayer_68796786147889
// MI455X (gfx1250) — compile-verified
//
#include <hip/hip_runtime.h>

typedef __attribute__((ext_vector_type(2))) float v2f;
typedef __attribute__((ext_vector_type(8))) float v8f;

#define N_NODES 50000
#define N_EDGES 800000
#define IN_F    256
#define HID     96
#define OUT_F   96

// ---------------------------------------------------------------------------
// Kernel 1: hlin = h @ W_lin.T + b_lin   via V_WMMA_F32_16X16X4_F32
// 192 threads = 6 waves; block handles 16 rows x 96 cols; K = 256 (64 wmmas)
// ---------------------------------------------------------------------------
__global__ __launch_bounds__(192) void k_linear(const float* __restrict__ h,
                                                const float* __restrict__ W_lin,
                                                const float* __restrict__ b_lin,
                                                float* __restrict__ hlin) {
    __shared__ float As[16][IN_F + 4];           // stride 260 -> conflict-free cols
    const int tid = threadIdx.x;
    const int row_base = blockIdx.x * 16;        // 50000/16 = 3125 exact

    // cooperative coalesced load of the 16x256 A tile (float4)
    for (int i = tid; i < 16 * (IN_F / 4); i += 192) {
        const int r  = i >> 6;                   // / 64
        const int c4 = i & 63;
        const float4 v = *(const float4*)&h[(size_t)(row_base + r) * IN_F + 4 * c4];
        *(float4*)&As[r][4 * c4] = v;
    }
    __syncthreads();

    const int wave = tid >> 5;
    const int lane = tid & 31;
    const int g = lane >> 4;                     // half-wave group
    const int t = lane & 15;
    const int colbase = wave * 16;

    v8f c = {};
    const float* wrow = &W_lin[(size_t)(colbase + t) * IN_F + 2 * g]; // B[k][n]=W[n][k]
    const float* arow = &As[t][2 * g];
    for (int kk = 0; kk < IN_F / 4; ++kk) {
        const float2 af2 = *(const float2*)&arow[4 * kk];   // A[t][4kk+2g .. +1]
        const float2 bf2 = *(const float2*)&wrow[4 * kk];   // B[4kk+2g .. +1][t]
        v2f a; a.x = af2.x; a.y = af2.y;
        v2f b; b.x = bf2.x; b.y = bf2.y;
        c = __builtin_amdgcn_wmma_f32_16x16x4_f32(false, a, false, b,
                                                  (short)0, c, false, false);
    }
    const float bias = b_lin[colbase + t];
#pragma unroll
    for (int v = 0; v < 8; ++v) {                // C reg v -> row v+8g, col t
        hlin[(size_t)(row_base + v + 8 * g) * HID + colbase + t] = c[v] + bias;
    }
}

// ---------------------------------------------------------------------------
// Kernel 2: zero the accumulators (workspace is poisoned by the harness)
// ---------------------------------------------------------------------------
__global__ void k_zero(float4* __restrict__ p, int n4) {
    const int i = blockIdx.x * blockDim.x + threadIdx.x;
    if (i < n4) p[i] = make_float4(0.f, 0.f, 0.f, 0.f);
}

// ---------------------------------------------------------------------------
// Kernel 3: edge scatter  acc[dst] += hlin[src] * norm[src]
// one wave32 per edge, 96 features = 3 coalesced 128B lines + 3 fp32 atomics
// ---------------------------------------------------------------------------
__global__ __launch_bounds__(256) void k_scatter(const float* __restrict__ hlin,
                                                 const float* __restrict__ norm,
                                                 const int* __restrict__ src,
                                                 const int* __restrict__ dst,
                                                 float* __restrict__ acc) {
    const int wave = threadIdx.x >> 5;
    const int lane = threadIdx.x & 31;
    const int e = blockIdx.x * 8 + wave;         // 800000/8 = 100000 exact
    const int s = src[e];
    const int d = dst[e];
    const float ns = norm[s];
    const size_t sb = (size_t)s * HID;
    const size_t db = (size_t)d * HID;
#pragma unroll
    for (int k = 0; k < 3; ++k) {
        const int f = lane + 32 * k;
        unsafeAtomicAdd(&acc[db + f], hlin[sb + f] * ns);   // global_atomic_add_f32
    }
}

// ---------------------------------------------------------------------------
// Kernel 4: h1/h2 = acc*norm, attention weights, combine, out = hatt@W_fc.T+b
// ---------------------------------------------------------------------------
__global__ __launch_bounds__(192) void k_combine(const float* __restrict__ hlin,
                                                 const float* __restrict__ acc1,
                                                 const float* __restrict__ acc2,
                                                 const float* __restrict__ norm1,
                                                 const float* __restrict__ norm2,
                                                 const float* __restrict__ W_al,
                                                 const float* __restrict__ b_al,
                                                 const float* __restrict__ W_ar,
                                                 const float* __restrict__ b_ar,
                                                 const float* __restrict__ W_fc,
                                                 const float* __restrict__ b_fc,
                                                 float* __restrict__ out) {
    __shared__ float Lh[16][HID + 4];
    __shared__ float L1[16][HID + 4];
    __shared__ float L2[16][HID + 4];
    __shared__ float w1s[16], w2s[16];

    const int tid = threadIdx.x;
    const int base = blockIdx.x * 16;

    for (int i = tid; i < 16 * HID; i += 192) {
        const int r = i / HID;
        const int cc = i - r * HID;
        const size_t idx = (size_t)(base + r) * HID + cc;
        Lh[r][cc] = hlin[idx];
        L1[r][cc] = acc1[idx] * norm1[base + r];
        L2[r][cc] = acc2[idx] * norm2[base + r];
    }
    __syncthreads();

    if (tid < 16) {                              // per-row attention scalars
        float ai = b_al[0], aj1 = b_ar[0], aj2 = b_ar[0];
        for (int k = 0; k < HID; ++k) {
            ai  += Lh[tid][k] * W_al[k];
            aj1 += L1[tid][k] * W_ar[k];
            aj2 += L2[tid][k] * W_ar[k];
        }
        float x1 = ai + aj1, x2 = ai + aj2;
        x1 = x1 >= 0.f ? x1 : 0.2f * x1;         // leaky relu
        x2 = x2 >= 0.f ? x2 : 0.2f * x2;
        const float a1 = fminf(fmaxf(expf(x1), -10.f), 10.f);
        const float a2 = fminf(fmaxf(expf(x2), -10.f), 10.f);
        const float denom = a1 + a2;
        w1s[tid] = a1 / denom;
        w2s[tid] = a2 / denom;
    }
    __syncthreads();

    for (int i = tid; i < 16 * HID; i += 192) {  // hatt in-place into L1
        const int r = i / HID;
        const int cc = i - r * HID;
        L1[r][cc] = w1s[r] * L1[r][cc] + w2s[r] * L2[r][cc];
    }
    __syncthreads();

    // WMMA: out tile = L1(16x96) @ W_fc.T(96x96), K = 96 -> 24 wmmas
    const int wave = tid >> 5;
    const int lane = tid & 31;
    const int g = lane >> 4;
    const int t = lane & 15;
    const int colbase = wave * 16;

    v8f c = {};
    const float* wrow = &W_fc[(size_t)(colbase + t) * HID + 2 * g];
    const float* arow = &L1[t][2 * g];
    for (int kk = 0; kk < HID / 4; ++kk) {
        const float2 af2 = *(const float2*)&arow[4 * kk];
        const float2 bf2 = *(const float2*)&wrow[4 * kk];
        v2f a; a.x = af2.x; a.y = af2.y;
        v2f b; b.x = bf2.x; b.y = bf2.y;
        c = __builtin_amdgcn_wmma_f32_16x16x4_f32(false, a, false, b,
                                                  (short)0, c, false, false);
    }
    const float bias = b_fc[colbase + t];
#pragma unroll
    for (int v = 0; v < 8; ++v) {
        out[(size_t)(base + v + 8 * g) * OUT_F + colbase + t] = c[v] + bias;
    }
}

// ---------------------------------------------------------------------------
extern "C" void kernel_launch(void* const* d_in, const int* in_sizes, int n_in,
                              void* d_out, int out_size, void* d_ws, size_t ws_size,
                              hipStream_t stream) {
    const float* h     = (const float*)d_in[0];
    const float* norm1 = (const float*)d_in[1];
    const float* norm2 = (const float*)d_in[2];
    const int*   src1  = (const int*)d_in[3];
    const int*   dst1  = (const int*)d_in[4];
    const int*   src2  = (const int*)d_in[5];
    const int*   dst2  = (const int*)d_in[6];
    const float* W_lin = (const float*)d_in[7];
    const float* b_lin = (const float*)d_in[8];
    const float* W_fc  = (const float*)d_in[9];
    const float* b_fc  = (const float*)d_in[10];
    const float* W_al  = (const float*)d_in[11];
    const float* b_al  = (const float*)d_in[12];
    const float* W_ar  = (const float*)d_in[13];
    const float* b_ar  = (const float*)d_in[14];
    float* out = (float*)d_out;

    float* hlin = (float*)d_ws;                       // [N, 96]
    float* acc1 = hlin + (size_t)N_NODES * HID;       // [N, 96]
    float* acc2 = acc1 + (size_t)N_NODES * HID;       // [N, 96]
    if (ws_size < 3 * (size_t)N_NODES * HID * sizeof(float)) return;

    k_linear<<<N_NODES / 16, 192, 0, stream>>>(h, W_lin, b_lin, hlin);

    const int n4 = 2 * N_NODES * HID / 4;             // zero acc1+acc2 (contiguous)
    k_zero<<<(n4 + 255) / 256, 256, 0, stream>>>((float4*)acc1, n4);

    k_scatter<<<N_EDGES / 8, 256, 0, stream>>>(hlin, norm1, src1, dst1, acc1);
    k_scatter<<<N_EDGES / 8, 256, 0, stream>>>(hlin, norm2, src2, dst2, acc2);

    k_combine<<<N_NODES / 16, 192, 0, stream>>>(hlin, acc1, acc2, norm1, norm2,
                                                W_al, b_al, W_ar, b_ar,
                                                W_fc, b_fc, out);
}